// StochasticAssociativeCuBaLIF_5789615915467
// MI455X (gfx1250) — compile-verified
//
#include <hip/hip_runtime.h>
#include <stdint.h>

#define B_SZ 16
#define T_SZ 2048
#define H_SZ 512
#define NSEG 8
#define SEGLEN 256           // T_SZ / NSEG
#define CHUNKS_PER_SEG 16    // SEGLEN / 16
#define HALF_N 8388608u      // (B*T*H)/2

typedef float v2f __attribute__((ext_vector_type(2)));
typedef float v8f __attribute__((ext_vector_type(8)));

__device__ __forceinline__ float clip01(float v){ return fminf(fmaxf(v, 0.f), 1.f); }

// a^e for e in [0,15]
__device__ __forceinline__ float powi16(float a, int e){
  float r = 1.f, p = a;
  #pragma unroll
  for (int i = 0; i < 4; ++i){ r = (e & 1) ? r * p : r; p *= p; e >>= 1; }
  return r;
}

// ---- threefry2x32-20, JAX key/rotation schedule ----
__device__ __forceinline__ unsigned rotl32(unsigned x, int n){ return (x << n) | (x >> (32 - n)); }
__device__ __forceinline__ void tf4(unsigned &x0, unsigned &x1, int r0, int r1, int r2, int r3){
  x0 += x1; x1 = rotl32(x1, r0); x1 ^= x0;
  x0 += x1; x1 = rotl32(x1, r1); x1 ^= x0;
  x0 += x1; x1 = rotl32(x1, r2); x1 ^= x0;
  x0 += x1; x1 = rotl32(x1, r3); x1 ^= x0;
}
__device__ __forceinline__ void threefry2x32(unsigned k0, unsigned k1,
                                             unsigned c0, unsigned c1,
                                             unsigned &o0, unsigned &o1){
  unsigned k2 = k0 ^ k1 ^ 0x1BD11BDAu;
  unsigned x0 = c0 + k0, x1 = c1 + k1;
  tf4(x0, x1, 13, 15, 26, 6);   x0 += k1; x1 += k2 + 1u;
  tf4(x0, x1, 17, 29, 16, 24);  x0 += k2; x1 += k0 + 2u;
  tf4(x0, x1, 13, 15, 26, 6);   x0 += k0; x1 += k1 + 3u;
  tf4(x0, x1, 17, 29, 16, 24);  x0 += k1; x1 += k2 + 4u;
  tf4(x0, x1, 13, 15, 26, 6);   x0 += k2; x1 += k0 + 5u;
  o0 = x0; o1 = x1;
}

// ---------- Phase 1: per-(b,h,segment) contribution from zero state ----------
__global__ __launch_bounds__(256) void cubalif_phase1(
    const float* __restrict__ u, const float* __restrict__ alpha,
    const float* __restrict__ beta, float* __restrict__ contrib)
{
  int gid = blockIdx.x * 256 + threadIdx.x;     // 65536 threads
  int h   = gid & (H_SZ - 1);
  int seg = (gid >> 9) & (NSEG - 1);
  int b   = gid >> 12;
  float a  = clip01(alpha[h]);
  float bb = clip01(beta[h]);
  float x = 0.f, V = 0.f;
  size_t base = ((size_t)b * T_SZ + (size_t)seg * SEGLEN) * H_SZ + h;
  for (int t = 0; t < SEGLEN; ++t){
    float uu = u[base + (size_t)t * H_SZ];
    x = fmaf(a, x, uu);
    V = fmaf(bb, V, x);
  }
  int ci = ((seg * 16 + b) * H_SZ + h) * 2;
  contrib[ci] = x; contrib[ci + 1] = V;
}

// ---------- Phase 2: prefix over 8 segments via closed-form M^256 ----------
__global__ __launch_bounds__(256) void cubalif_phase2(
    const float* __restrict__ alpha, const float* __restrict__ beta,
    const float* __restrict__ contrib, float* __restrict__ carry)
{
  int gid = blockIdx.x * 256 + threadIdx.x;     // 8192 threads
  int h = gid & (H_SZ - 1);
  int b = gid >> 9;
  float a  = clip01(alpha[h]);
  float bb = clip01(beta[h]);
  float aL = a, bL = bb;
  #pragma unroll
  for (int i = 0; i < 8; ++i){ aL *= aL; bL *= bL; }   // a^256, b^256
  float d  = a - bb;
  float cL = (fabsf(d) > 1e-5f) ? a * (aL - bL) / d : 256.f * aL;  // x0 coeff in V_L
  float x = 0.f, V = 0.f;
  for (int seg = 0; seg < NSEG; ++seg){
    int ci = ((seg * 16 + b) * H_SZ + h) * 2;
    carry[ci] = x; carry[ci + 1] = V;
    float cx = contrib[ci], cv = contrib[ci + 1];
    float xn = fmaf(aL, x, cx);
    float Vn = fmaf(cL, x, fmaf(bL, V, cv));
    x = xn; V = Vn;
  }
}

// ---------- Phase 3: WMMA chunked scan + spike + threefry bernoulli ----------
__global__ __launch_bounds__(256) void cubalif_phase3(
    const float* __restrict__ u, const float* __restrict__ alpha,
    const float* __restrict__ beta, const float* __restrict__ carry,
    const int* __restrict__ seedp, float* __restrict__ out)
{
  __shared__ __align__(16) float tile[2][16 * 16 * 8];   // [buf][(t*16+b)*8+h']
  const int tid  = threadIdx.x;
  const int w    = tid >> 5;          // wave -> channel within block
  const int lane = tid & 31;
  const int blk  = blockIdx.x;
  const int h0   = (blk & 63) << 3;   // 64 h-groups of 8
  const int seg  = blk >> 6;          // 8 segments
  const int h    = h0 + w;
  const int m    = lane & 15;         // A-row (time) / B-col (batch) lane index
  const int kh   = lane >> 4;

  const float a  = clip01(alpha[h]);
  const float bb = clip01(beta[h]);

  // A operands: lower-triangular Toeplitz of decay powers, 4 K-groups of 4
  v2f Aa[4], Ab[4];
  #pragma unroll
  for (int g = 0; g < 4; ++g){
    #pragma unroll
    for (int j = 0; j < 2; ++j){
      int s = 4 * g + 2 * kh + j;
      Aa[g][j] = (s <= m) ? powi16(a,  m - s) : 0.f;
      Ab[g][j] = (s <= m) ? powi16(bb, m - s) : 0.f;
    }
  }
  // carry outer-product factors: decay^(row+1), row = i + 8*kh
  float fa[8], fb[8];
  fa[0] = kh ? powi16(a, 9) : a;
  fb[0] = kh ? powi16(bb, 9) : bb;
  #pragma unroll
  for (int i = 1; i < 8; ++i){ fa[i] = fa[i - 1] * a; fb[i] = fb[i - 1] * bb; }

  // segment carry-in for (batch=m, h)
  int ci = ((seg * 16 + m) * H_SZ + h) * 2;
  float x_in = carry[ci];
  float V_in = carry[ci + 1];

  const unsigned k0 = 0u;
  const unsigned k1 = (unsigned)seedp[0];

  // async staging: thread q stages row (b=q>>4, t_local=q&15), 8 floats (2x b128)
  const int q_b  = tid >> 4;
  const int q_tl = tid & 15;
  const unsigned lds_row0 = (unsigned)(uintptr_t)&tile[0][(q_tl * 16 + q_b) * 8];
  const unsigned lds_row1 = (unsigned)(uintptr_t)&tile[1][(q_tl * 16 + q_b) * 8];
  const int t_base = seg * SEGLEN;

  { // prefetch chunk 0 -> buffer 0
    unsigned voff = (unsigned)(((q_b * T_SZ + t_base + q_tl) * H_SZ + h0) * 4);
    asm volatile("global_load_async_to_lds_b128 %0, %1, %2"
                 :: "v"(lds_row0), "v"(voff), "s"(u) : "memory");
    asm volatile("global_load_async_to_lds_b128 %0, %1, %2"
                 :: "v"(lds_row0 + 16u), "v"(voff + 16u), "s"(u) : "memory");
  }

  for (int c = 0; c < CHUNKS_PER_SEG; ++c){
    const int t0 = t_base + c * 16;
    if (c + 1 < CHUNKS_PER_SEG){   // prefetch next chunk into other buffer
      unsigned ldsr = ((c + 1) & 1) ? lds_row1 : lds_row0;
      unsigned voff = (unsigned)(((q_b * T_SZ + t0 + 16 + q_tl) * H_SZ + h0) * 4);
      asm volatile("global_load_async_to_lds_b128 %0, %1, %2"
                   :: "v"(ldsr), "v"(voff), "s"(u) : "memory");
      asm volatile("global_load_async_to_lds_b128 %0, %1, %2"
                   :: "v"(ldsr + 16u), "v"(voff + 16u), "s"(u) : "memory");
      asm volatile("s_wait_asynccnt 2" ::: "memory");   // current buffer done
    } else {
      asm volatile("s_wait_asynccnt 0" ::: "memory");
    }
    __syncthreads();

    const float* tb = tile[c & 1];

    // x tile: D = sum_g L(a)[g] * U[g] + outer(a^(t+1), x_in)
    v8f x;
    #pragma unroll
    for (int i = 0; i < 8; ++i) x[i] = fa[i] * x_in;
    #pragma unroll
    for (int g = 0; g < 4; ++g){
      int sb = 4 * g + 2 * kh;
      v2f Bg;
      Bg[0] = tb[((sb    ) * 16 + m) * 8 + w];
      Bg[1] = tb[((sb + 1) * 16 + m) * 8 + w];
      x = __builtin_amdgcn_wmma_f32_16x16x4_f32(false, Aa[g], false, Bg,
                                                (short)0, x, false, false);
    }

    // re-lay x (C/D layout) into B layout for second WMMA via half-wave exchange
    float xv[8], sx[8];
    #pragma unroll
    for (int i = 0; i < 8; ++i){ xv[i] = x[i]; sx[i] = __shfl_xor(xv[i], 16, 32); }
    x_in = kh ? xv[7] : sx[7];                 // x[15][m] -> next chunk carry

    v2f B2[4];
    B2[0][0] = kh ? sx[2] : xv[0];  B2[0][1] = kh ? sx[3] : xv[1];
    B2[1][0] = kh ? sx[6] : xv[4];  B2[1][1] = kh ? sx[7] : xv[5];
    B2[2][0] = kh ? xv[2] : sx[0];  B2[2][1] = kh ? xv[3] : sx[1];
    B2[3][0] = kh ? xv[6] : sx[4];  B2[3][1] = kh ? xv[7] : sx[5];

    v8f V;
    #pragma unroll
    for (int i = 0; i < 8; ++i) V[i] = fb[i] * V_in;
    #pragma unroll
    for (int g = 0; g < 4; ++g)
      V = __builtin_amdgcn_wmma_f32_16x16x4_f32(false, Ab[g], false, B2[g],
                                                (short)0, V, false, false);

    float v7p = __shfl_xor((float)V[7], 16, 32);
    V_in = kh ? (float)V[7] : v7p;

    // spike probability + bernoulli + store (rows M = i + 8*kh)
    #pragma unroll
    for (int i = 0; i < 8; ++i){
      int t = t0 + i + 8 * kh;
      unsigned idx = (unsigned)((m * T_SZ + t) * H_SZ + h);
      float U   = (float)V[i] - 1.0f;
      float sig = 1.f / (1.f + __expf(-100.f * U));
      float pr  = fmaxf(__cosf(6.28318530717958647692f * U) * sig, 0.f);
      unsigned pi = (idx >= HALF_N) ? (idx - HALF_N) : idx;
      unsigned o0, o1;
      threefry2x32(k0, k1, pi, pi + HALF_N, o0, o1);
      unsigned bits = (idx >= HALF_N) ? o1 : o0;
      float r = __uint_as_float((bits >> 9) | 0x3f800000u) - 1.0f;
      out[idx] = (r < pr) ? 1.f : 0.f;
    }
    __syncthreads();   // buffer reuse fence before next prefetch
  }
}

extern "C" void kernel_launch(void* const* d_in, const int* in_sizes, int n_in,
                              void* d_out, int out_size, void* d_ws, size_t ws_size,
                              hipStream_t stream) {
  const float* u     = (const float*)d_in[0];
  const float* alpha = (const float*)d_in[1];
  const float* beta  = (const float*)d_in[2];
  const int*   seed  = (const int*)d_in[3];
  float* out = (float*)d_out;
  float* contrib = (float*)d_ws;                        // 8*16*512*2 f32 = 512 KB
  float* carry   = contrib + NSEG * B_SZ * H_SZ * 2;    // +512 KB
  cubalif_phase1<<<(NSEG * B_SZ * H_SZ) / 256, 256, 0, stream>>>(u, alpha, beta, contrib);
  cubalif_phase2<<<(B_SZ * H_SZ) / 256, 256, 0, stream>>>(alpha, beta, contrib, carry);
  cubalif_phase3<<<(H_SZ / 8) * NSEG, 256, 0, stream>>>(u, alpha, beta, carry, seed, out);
  (void)in_sizes; (void)n_in; (void)out_size; (void)ws_size;
}